// DownsampleBlock_82617990906063
// MI455X (gfx1250) — compile-verified
//
#include <hip/hip_runtime.h>
#include <hip/hip_bf16.h>

// ---------------------------------------------------------------------------
// Problem constants (fixed by setup_inputs in the reference)
// ---------------------------------------------------------------------------
#define NPTS   65536          // input points
#define MOUT   16384          // NPTS / DOWN_SAMPLE
#define KNN_K  16
#define CHN    128            // in_dim == out_dim == 128

typedef __attribute__((ext_vector_type(2))) float v2f;
typedef __attribute__((ext_vector_type(8))) float v8f;
typedef __attribute__((ext_vector_type(4))) int   v4i;

// CDNA5 async-to-LDS path (ASYNCcnt-tracked). Guarded so the build cannot
// regress if this toolchain doesn't expose the builtins.
#if defined(__has_builtin)
#if __has_builtin(__builtin_amdgcn_global_load_async_to_lds_b128) && \
    __has_builtin(__builtin_amdgcn_s_wait_asynccnt)
#define HAVE_ASYNC_LDS 1
#endif
#endif
#ifndef HAVE_ASYNC_LDS
#define HAVE_ASYNC_LDS 0
#endif

// ---------------------------------------------------------------------------
// 1) Farthest point sampling: single persistent workgroup (1024 threads).
//    min_d lives in a global scratch slab (L2 resident); argmax via LDS tree
//    with lowest-index tie-breaking to match jnp.argmax.
// ---------------------------------------------------------------------------
__global__ __launch_bounds__(1024) void fps_kernel(
    const float* __restrict__ pos, float* __restrict__ mind,
    int* __restrict__ index)
{
    __shared__ float rv[1024];
    __shared__ int   ri[1024];
    const int t = threadIdx.x;

    #pragma unroll
    for (int i = 0; i < NPTS / 1024; ++i) mind[t + (i << 10)] = 1.0e30f;
    if (t == 0) index[0] = 0;
    __syncthreads();

    int last = 0;
    for (int s = 1; s < MOUT; ++s) {
        const float lx = pos[last * 3 + 0];
        const float ly = pos[last * 3 + 1];
        const float lz = pos[last * 3 + 2];

        float best = -1.0f;
        int   bidx = 0;
        #pragma unroll 4
        for (int i = 0; i < NPTS / 1024; ++i) {
            const int p = t + (i << 10);
            const float dx = pos[p * 3 + 0] - lx;
            const float dy = pos[p * 3 + 1] - ly;
            const float dz = pos[p * 3 + 2] - lz;
            float d  = dx * dx + dy * dy + dz * dz;
            float md = mind[p];
            md = fminf(md, d);
            mind[p] = md;
            if (md > best) { best = md; bidx = p; }   // ascending p scan keeps first max
        }
        rv[t] = best; ri[t] = bidx;
        __syncthreads();
        for (int off = 512; off > 0; off >>= 1) {
            if (t < off) {
                const float v2 = rv[t + off];
                const int   i2 = ri[t + off];
                if (v2 > rv[t] || (v2 == rv[t] && i2 < ri[t])) { rv[t] = v2; ri[t] = i2; }
            }
            __syncthreads();
        }
        last = ri[0];
        if (t == 0) index[s] = last;
        __syncthreads();   // protect rv/ri before next iteration overwrites
    }
}

// ---------------------------------------------------------------------------
// 2) Gather sampled centers: qpos (scratch) + pos/batch tuple outputs.
// ---------------------------------------------------------------------------
__global__ __launch_bounds__(256) void gather_kernel(
    const float* __restrict__ pos, const int* __restrict__ batch,
    const int* __restrict__ index, float* __restrict__ qpos,
    float* __restrict__ pos_out, int* __restrict__ batch_out)
{
    const int m = blockIdx.x * blockDim.x + threadIdx.x;
    if (m >= MOUT) return;
    const int idx = index[m];
    const float px = pos[idx * 3 + 0];
    const float py = pos[idx * 3 + 1];
    const float pz = pos[idx * 3 + 2];
    qpos[m * 3 + 0] = px;  pos_out[m * 3 + 0] = px;
    qpos[m * 3 + 1] = py;  pos_out[m * 3 + 1] = py;
    qpos[m * 3 + 2] = pz;  pos_out[m * 3 + 2] = pz;
    batch_out[m] = batch[idx];
}

// ---------------------------------------------------------------------------
// 3) h = relu(x @ W^T + b) with V_WMMA_F32_16X16X4_F32 (exact fp32 path; the
//    GEMM is memory bound at ~64MB / 23.3TB/s so fp32 WMMA costs nothing).
//    One wave per 16x16 output tile; 8 waves/block cover all 128 channels of
//    one 16-row stripe. A-frag (x) and B-frag (rows of W) share the same
//    contiguous float2 access pattern per the ISA 16x4 layout.
// ---------------------------------------------------------------------------
__global__ __launch_bounds__(256) void gemm_relu_kernel(
    const float* __restrict__ x, const float* __restrict__ W,
    const float* __restrict__ bias, float* __restrict__ h)
{
    const int wave = threadIdx.x >> 5;     // 0..7  -> channel tile
    const int lane = threadIdx.x & 31;
    const int half = lane >> 4;            // 0: K={0,1}, 1: K={2,3}
    const int l15  = lane & 15;
    const int row0 = blockIdx.x * 16;      // 4096 blocks
    const int o0   = wave * 16;

    const float* xrow = x + (size_t)(row0 + l15) * CHN + half * 2;
    const float* wrow = W + (size_t)(o0   + l15) * CHN + half * 2;

    v8f c = {};
    #pragma unroll 8
    for (int k0 = 0; k0 < CHN; k0 += 4) {
        const v2f a  = *(const v2f*)(xrow + k0);
        const v2f bb = *(const v2f*)(wrow + k0);
        // (neg_a, A, neg_b, B, c_mod, C, reuse_a, reuse_b)
        c = __builtin_amdgcn_wmma_f32_16x16x4_f32(
                false, a, false, bb, (short)0, c, false, false);
    }

    const float bv = bias[o0 + l15];
    float* hp = h + (size_t)(row0 + half * 8) * CHN + o0 + l15;
    #pragma unroll
    for (int r = 0; r < 8; ++r) {          // VGPR r -> row r (lanes 0-15) / r+8
        const float v = c[r] + bv;
        hp[(size_t)r * CHN] = fmaxf(v, 0.0f);
    }
}

// ---------------------------------------------------------------------------
// 4) KNN: 8 waves/block, one query center per wave. pos double-buffered
//    through LDS in 2048-point AoS chunks via CDNA5 async-to-LDS B128 copies
//    (ASYNCcnt) so the L2->LDS stream overlaps the selection ALU work.
//    Register-resident sorted top-16 per lane, then 32-way tournament merge
//    via wave32 shfl_xor with (dist, idx) tie-break (jax.lax.top_k order).
// ---------------------------------------------------------------------------
#define KNN_CHUNK 2048
#define KNN_CD    (KNN_CHUNK * 3)          // dwords per chunk (6144 = 24KB)

__device__ __forceinline__ void knn_stage_async(const float* __restrict__ gsrc,
                                                float* lds_dst, int t)
{
#if HAVE_ASYNC_LDS
    #pragma unroll
    for (int k = 0; k < KNN_CD / 4 / 256; ++k) {     // 6 x B128 per thread
        const int d = (t + k * 256) * 4;             // dword index, 16B aligned
        __builtin_amdgcn_global_load_async_to_lds_b128(
            (__attribute__((address_space(1))) v4i*)(gsrc + d),
            (__attribute__((address_space(3))) v4i*)(lds_dst + d),
            0, 0);
    }
#else
    (void)gsrc; (void)lds_dst; (void)t;
#endif
}

__global__ __launch_bounds__(256) void knn_kernel(
    const float* __restrict__ pos, const float* __restrict__ qpos,
    int* __restrict__ nbrs)
{
    __shared__ float sp[2][KNN_CD];        // AoS x,y,z: stride-3 -> bank-conflict free

    const int t    = threadIdx.x;
    const int wave = t >> 5;
    const int lane = t & 31;
    const int center = blockIdx.x * 8 + wave;

    const float qx = qpos[center * 3 + 0];
    const float qy = qpos[center * 3 + 1];
    const float qz = qpos[center * 3 + 2];

    float nd[KNN_K];
    int   ni[KNN_K];
    #pragma unroll
    for (int j = 0; j < KNN_K; ++j) { nd[j] = 3.0e30f; ni[j] = 0x7fffffff; }

#if HAVE_ASYNC_LDS
    knn_stage_async(pos, sp[0], t);        // prologue: chunk 0 in flight
#endif
    int buf = 0;
    for (int c0 = 0; c0 < NPTS; c0 += KNN_CHUNK) {
#if HAVE_ASYNC_LDS
        __builtin_amdgcn_s_wait_asynccnt(0);   // own async copies landed in LDS
        __syncthreads();                       // -> whole buffer valid for block
        if (c0 + KNN_CHUNK < NPTS)             // overlap next copy with compute
            knn_stage_async(pos + (size_t)(c0 + KNN_CHUNK) * 3, sp[buf ^ 1], t);
#else
        __syncthreads();
        for (int i = t; i < KNN_CD; i += 256)
            sp[buf][i] = pos[(size_t)c0 * 3 + i];
        __syncthreads();
#endif
        const float* P = sp[buf];
        for (int i = lane; i < KNN_CHUNK; i += 32) {
            const float dx = P[i * 3 + 0] - qx;
            const float dy = P[i * 3 + 1] - qy;
            const float dz = P[i * 3 + 2] - qz;
            const float d  = dx * dx + dy * dy + dz * dz;
            if (d < nd[KNN_K - 1]) {       // rare after warm-up
                int j = KNN_K - 1;
                while (j > 0 && nd[j - 1] > d) {
                    nd[j] = nd[j - 1]; ni[j] = ni[j - 1]; --j;
                }
                nd[j] = d; ni[j] = c0 + i;
            }
        }
        buf ^= 1;
    }

    // Tournament merge of 32 sorted lists (butterfly allreduce per round).
    int ptr  = 0;
    int keep = 0;
    #pragma unroll
    for (int r = 0; r < KNN_K; ++r) {
        float v  = (ptr < KNN_K) ? nd[ptr] : 3.0e30f;
        int   vi = (ptr < KNN_K) ? ni[ptr] : 0x7fffffff;
        int   vl = lane;
        #pragma unroll
        for (int off = 16; off > 0; off >>= 1) {
            const float ov  = __shfl_xor(v,  off);
            const int   ovi = __shfl_xor(vi, off);
            const int   ovl = __shfl_xor(vl, off);
            if (ov < v || (ov == v && ovi < vi)) { v = ov; vi = ovi; vl = ovl; }
        }
        if (lane == vl) ++ptr;             // winner advances its list
        if (lane == r)  keep = vi;         // lane r holds r-th nearest
    }
    if (lane < KNN_K) nbrs[(size_t)center * KNN_K + lane] = keep;
}

// ---------------------------------------------------------------------------
// 5) out[m, :] = max_k h[nbrs[m, k], :]
// ---------------------------------------------------------------------------
__global__ __launch_bounds__(128) void maxgather_kernel(
    const float* __restrict__ h, const int* __restrict__ nbrs,
    float* __restrict__ out)
{
    const int m = blockIdx.x;
    const int c = threadIdx.x;
    float v = -3.0e30f;
    #pragma unroll
    for (int k = 0; k < KNN_K; ++k) {
        const int p = nbrs[(size_t)m * KNN_K + k];
        v = fmaxf(v, h[(size_t)p * CHN + c]);
    }
    out[(size_t)m * CHN + c] = v;
}

// ---------------------------------------------------------------------------
// Launch
// ---------------------------------------------------------------------------
extern "C" void kernel_launch(void* const* d_in, const int* in_sizes, int n_in,
                              void* d_out, int out_size, void* d_ws, size_t ws_size,
                              hipStream_t stream) {
    const float* x    = (const float*)d_in[0];   // [65536,128]
    const float* pos  = (const float*)d_in[1];   // [65536,3]
    const int*   bat  = (const int*)  d_in[2];   // [65536]
    const float* W    = (const float*)d_in[3];   // [128,128]
    const float* bias = (const float*)d_in[4];   // [128]

    char* ws = (char*)d_ws;
    int*   index = (int*)(ws + 0);                               //   64 KB
    int*   nbrs  = (int*)(ws + 65536);                           //    1 MB
    float* qpos  = (float*)(ws + 65536 + 1048576);               //  192 KB
    float* h     = (float*)(ws + 65536 + 1048576 + 196608);      //   32 MB
    float* mind  = (float*)(ws + 65536 + 1048576 + 196608 + 33554432); // 256 KB

    float* out     = (float*)d_out;                  // [16384,128]
    float* pos_out = out + (size_t)MOUT * CHN;       // [16384,3]
    int*   bat_out = (int*)(pos_out + (size_t)MOUT * 3); // [16384] int bits

    // Independent of FPS: matrix path first so it overlaps nothing serial.
    gemm_relu_kernel<<<NPTS / 16, 256, 0, stream>>>(x, W, bias, h);

    fps_kernel<<<1, 1024, 0, stream>>>(pos, mind, index);

    gather_kernel<<<MOUT / 256, 256, 0, stream>>>(pos, bat, index, qpos,
                                                  pos_out, bat_out);

    knn_kernel<<<MOUT / 8, 256, 0, stream>>>(pos, qpos, nbrs);

    maxgather_kernel<<<MOUT, 128, 0, stream>>>(h, nbrs, out);
}